// RootLayer_49916109914251
// MI455X (gfx1250) — compile-verified
//
#include <hip/hip_runtime.h>
#include <math.h>

typedef float v2f __attribute__((ext_vector_type(2)));
typedef float v8f __attribute__((ext_vector_type(8)));

#define DDIM 2048
#define CCLS 10
#define BROWS 8192
#define WCH 128            // chunk width (columns)
#define WPAD 132           // padded LDS row stride in floats (132 % 64 == 4 -> no bank conflicts)
#define NCHUNK (DDIM / WCH)  // 16
#define KKPC (WCH / 4)       // 32 wmma K-steps per chunk

// ---------------------------------------------------------------- zero ws
__global__ void zero_ws_kernel(float* __restrict__ ws) {
    int i = blockIdx.x * blockDim.x + threadIdx.x;   // grid covers 512*64 = 32768 floats
    ws[i] = 0.0f;
}

// ------------------------------------------- softmax(weight) -> packed B fragments
// sfrag layout: [kk=0..511][64 floats], where for element d = 4*kk + k of class c:
//   lane = c + 16*(k>>1), component = k&1  -> float index kk*64 + lane*2 + (k&1)
__global__ __launch_bounds__(256) void softmax_pack_kernel(const float* __restrict__ w,
                                                           float* __restrict__ sfrag) {
    __shared__ float red[256];
    const int c = blockIdx.x;
    const int tid = threadIdx.x;
    const float* wr = w + c * DDIM;

    float m = -3.4e38f;
    for (int d = tid; d < DDIM; d += 256) m = fmaxf(m, wr[d]);
    red[tid] = m; __syncthreads();
    for (int s = 128; s > 0; s >>= 1) { if (tid < s) red[tid] = fmaxf(red[tid], red[tid + s]); __syncthreads(); }
    m = red[0]; __syncthreads();

    float sum = 0.0f;
    for (int d = tid; d < DDIM; d += 256) sum += __expf(wr[d] - m);
    red[tid] = sum; __syncthreads();
    for (int s = 128; s > 0; s >>= 1) { if (tid < s) red[tid] += red[tid + s]; __syncthreads(); }
    const float inv = 1.0f / red[0];

    for (int d = tid; d < DDIM; d += 256) {
        float v = __expf(wr[d] - m) * inv;            // exp(log_softmax) = softmax
        int kk = d >> 2, k = d & 3;
        sfrag[kk * 64 + (c + 16 * (k >> 1)) * 2 + (k & 1)] = v;
    }
}

// ---------------------------------------------------------------- main fused kernel
__global__ __launch_bounds__(128) void lse_wmma_kernel(const float* __restrict__ x,
                                                       const float* __restrict__ sfrag,
                                                       float* __restrict__ out) {
    __shared__ float xlds[4][16 * WPAD];   // per-wave staged x tile chunk
    __shared__ float mrow[4][16];          // per-wave running row max
    __shared__ float cmax[4][16];          // per-wave chunk row max
    __shared__ float fsc[4][16];           // per-wave rescale factors

    const int lane = threadIdx.x & 31;
    const int wv   = threadIdx.x >> 5;
    const int tile = blockIdx.x * 4 + wv;  // 512 tiles of 16 rows
    const int r0   = tile * 16;
    const int rA   = lane & 15;            // A-matrix row / B-matrix class for this lane
    const int hh   = lane >> 4;            // K-half select per WMMA fragment layout

    float* xw = xlds[wv];
    float* mr = mrow[wv];
    float* cm = cmax[wv];
    float* fw = fsc[wv];

    if (lane < 16) mr[lane] = -3.0e38f;
    v8f acc = {0.f, 0.f, 0.f, 0.f, 0.f, 0.f, 0.f, 0.f};
    __syncthreads();

    for (int ch = 0; ch < NCHUNK; ++ch) {
        // ---- stage 16xWCH chunk into LDS (coalesced 512B/iter) + per-row chunk max
        const float* xg = x + (size_t)r0 * DDIM + ch * WCH + 4 * lane;
        #pragma unroll
        for (int i = 0; i < 16; ++i) {
            float4 v = *(const float4*)(xg + (size_t)i * DDIM);
            *(float4*)(xw + i * WPAD + 4 * lane) = v;
            float lm = fmaxf(fmaxf(v.x, v.y), fmaxf(v.z, v.w));
            #pragma unroll
            for (int off = 16; off > 0; off >>= 1)
                lm = fmaxf(lm, __shfl_xor(lm, off, 32));
            if (lane == i) cm[i] = lm;     // all lanes hold row i's max here
        }
        __syncthreads();

        // ---- online logsumexp rescale
        float mo  = mr[rA];
        float cmx = cm[rA];
        float mn  = fmaxf(mo, cmx);        // new running max for row rA
        __syncthreads();
        if (lane < 16) { mr[lane] = mn; fw[lane] = __expf(mo - mn); }
        __syncthreads();
        {
            const int base = (lane < 16) ? 0 : 8;   // C layout: VGPR v -> row v (lanes 0-15) / v+8
            #pragma unroll
            for (int v = 0; v < 8; ++v) acc[v] *= fw[base + v];
        }

        // ---- WMMA over chunk: D += exp(x - m) * softmax(w)^T
        const float* ap = xw + rA * WPAD + 2 * hh;                    // A frag: row rA, K half hh
        const float* bp = sfrag + (size_t)ch * KKPC * 64 + 2 * lane;  // pre-packed B frags
        #pragma unroll 4
        for (int kk = 0; kk < KKPC; ++kk) {
            float2 a = *(const float2*)(ap + 4 * kk);   // ds_load_b64, conflict-free
            float2 b = *(const float2*)(bp + 64 * kk);  // global_load_b64, coalesced, L2-hot
            v2f av; av.x = __expf(a.x - mn); av.y = __expf(a.y - mn);
            v2f bv; bv.x = b.x; bv.y = b.y;
            acc = __builtin_amdgcn_wmma_f32_16x16x4_f32(false, av, false, bv,
                                                        (short)0, acc, false, false);
        }
        __syncthreads();   // keep block waves in lockstep for next chunk's LDS reuse
    }

    // ---- epilogue: out[b,c] = m_row + log(acc)
    const int base = (lane < 16) ? 0 : 8;
    if (rA < CCLS) {
        #pragma unroll
        for (int v = 0; v < 8; ++v) {
            float mf = mr[base + v];
            out[(size_t)(r0 + base + v) * CCLS + rA] = mf + __logf(acc[v]);
        }
    }
}

// ---------------------------------------------------------------- launcher
extern "C" void kernel_launch(void* const* d_in, const int* in_sizes, int n_in,
                              void* d_out, int out_size, void* d_ws, size_t ws_size,
                              hipStream_t stream) {
    const float* x = (const float*)d_in[0];   // [8192, 64, 32] fp32
    const float* w = (const float*)d_in[1];   // [10, 2048] fp32
    float* out   = (float*)d_out;             // [8192, 10] fp32
    float* sfrag = (float*)d_ws;              // 512*64 floats = 128 KB packed B fragments

    zero_ws_kernel<<<128, 256, 0, stream>>>(sfrag);
    softmax_pack_kernel<<<CCLS, 256, 0, stream>>>(w, sfrag);
    lse_wmma_kernel<<<BROWS / 64, 128, 0, stream>>>(x, sfrag, out);
}